// ValuesAroundPump_24721831756549
// MI455X (gfx1250) — compile-verified
//
#include <hip/hip_runtime.h>

// ---------------------------------------------------------------------------
// ValuesAroundPump: per-sample 5x5 window mean over channels [2:6), broadcast
// to the full (H,W) plane.  B=64, H=W=512, C=6, RADIUS=2.
//
// Phase 1: WMMA f32 16x16x4 reduction. One wave handles 4 samples:
//   D row M = (sample_in_wave * 4 + channel), K = window-position chunk of 4,
//   B = all-ones (layout independent), 7 chained accumulates cover p=0..24
//   (p=25..27 zero-padded).  D[M][n] is identical for all n; lanes 0 and 16
//   hold column 0 of rows 0..7 / 8..15 in their 8 accumulator VGPRs.
// Phase 2: non-temporal b128 broadcast fill (256 MB > 192 MB L2 -> NT).
// ---------------------------------------------------------------------------

typedef float v2f __attribute__((ext_vector_type(2)));
typedef float v8f __attribute__((ext_vector_type(8)));
typedef float vf4 __attribute__((ext_vector_type(4)));

#define RADIUS 2
#define WIN 5
#define WIN2 25          // window positions
#define HH 512
#define WW 512
#define CC 6
#define NB 64            // batch
#define OUTC 4           // output channels (C-2)

// ---------------- Phase 1: per-sample window means via WMMA ----------------
// grid: 16 blocks of 32 threads (1 wave); block b handles samples 4b..4b+3.
__global__ __launch_bounds__(32)
void pump_means_wmma(const float* __restrict__ fields,
                     const int*   __restrict__ pump,
                     float*       __restrict__ means) {
  const int lane = threadIdx.x;        // 0..31
  const int M    = lane & 15;          // D-matrix row this lane feeds for A
  const int hi   = lane >> 4;          // 0 -> K={0,1}, 1 -> K={2,3} (ISA A layout)
  const int sl   = M >> 2;             // sample-in-wave 0..3
  const int ch   = M & 3;              // output channel 0..3 (fields channel 2+ch)
  const int s    = blockIdx.x * 4 + sl;

  const int py = pump[2 * s + 0];
  const int px = pump[2 * s + 1];
  const long sbase = (long)s * HH * WW * CC;

  v8f acc = {};                        // C/D accumulator (16x16 f32, 8 VGPRs)
  v2f bones;                           // B = 4x16 all-ones: splat is layout-safe
  bones.x = 1.0f;
  bones.y = 1.0f;

#pragma unroll
  for (int j = 0; j < 7; ++j) {        // 7 * K4 = 28 >= 25 positions
    const int p0 = 4 * j + hi * 2;     // K index this lane supplies in VGPR0
    const int p1 = p0 + 1;             // ... and VGPR1
    v2f a;
    a.x = 0.0f;
    a.y = 0.0f;
    if (p0 < WIN2) {
      const int dy = p0 / WIN, dx = p0 % WIN;
      a.x = fields[sbase + ((long)(py - RADIUS + dy) * WW + (px - RADIUS + dx)) * CC + 2 + ch];
    }
    if (p1 < WIN2) {
      const int dy = p1 / WIN, dx = p1 % WIN;
      a.y = fields[sbase + ((long)(py - RADIUS + dy) * WW + (px - RADIUS + dx)) * CC + 2 + ch];
    }
    // D = A(16x4) * ones(4x16) + C : row-sums accumulate across chunks.
    acc = __builtin_amdgcn_wmma_f32_16x16x4_f32(
        /*neg_a=*/false, a, /*neg_b=*/false, bones,
        /*c_mod=*/(short)0, acc, /*reuse_a=*/false, /*reuse_b=*/false);
  }

  // Extract column 0: lane 0 holds rows 0..7 (VGPR v -> M=v),
  // lane 16 holds rows 8..15 (VGPR v -> M=v+8).
  if ((lane & 15) == 0) {
#pragma unroll
    for (int v = 0; v < 8; ++v) {
      const int row = v + hi * 8;                      // row = sl*4 + ch
      means[blockIdx.x * 16 + row] = acc[v] * (1.0f / 25.0f);
    }
  }
}

// ---------------- Phase 2: non-temporal broadcast fill ---------------------
// out is (B, H, W, 4) f32 -> one vf4 per (h,w).  262144 vf4 per sample.
// grid (64 blocks/sample, 64 samples), 256 threads, 16 vf4 stores per thread.
__global__ __launch_bounds__(256)
void broadcast_fill(const float* __restrict__ means, vf4* __restrict__ out) {
  const int s = blockIdx.y;
  vf4 m;
  m.x = means[4 * s + 0];
  m.y = means[4 * s + 1];
  m.z = means[4 * s + 2];
  m.w = means[4 * s + 3];

  const size_t base = (size_t)s * ((size_t)HH * WW)
                    + (size_t)blockIdx.x * (256 * 16)
                    + threadIdx.x;
#pragma unroll
  for (int i = 0; i < 16; ++i) {
    // Streaming store: 256 MB output exceeds 192 MB L2 -> NT to avoid churn.
    __builtin_nontemporal_store(m, out + base + (size_t)i * 256);
  }
}

extern "C" void kernel_launch(void* const* d_in, const int* in_sizes, int n_in,
                              void* d_out, int out_size, void* d_ws, size_t ws_size,
                              hipStream_t stream) {
  const float* fields = (const float*)d_in[0];
  const int*   pump   = (const int*)d_in[1];
  float*       means  = (float*)d_ws;         // NB * OUTC = 256 floats scratch
  vf4*         out    = (vf4*)d_out;

  pump_means_wmma<<<dim3(NB / 4), dim3(32), 0, stream>>>(fields, pump, means);
  broadcast_fill<<<dim3(64, NB), dim3(256), 0, stream>>>(means, out);
}